// YoloLossV8_44676249813103
// MI455X (gfx1250) — compile-verified
//
#include <hip/hip_runtime.h>
#include <math.h>

// ---------------- problem constants ----------------
#define NB      32
#define NA      8400
#define BANCH   (NB * NA)          // 268800
#define NC      80
#define REGMAX  16
#define EPSF    1e-7f

#define BDIM    128
#define WAVES   (BDIM / 32)
#define NBLOCKS (BANCH / BDIM)     // 2100, exact

typedef __attribute__((ext_vector_type(2))) float v2f;
typedef __attribute__((ext_vector_type(8))) float v8f;

// fast transcendentals -> single v_exp_f32 / v_log_f32 TRANS ops
__device__ __forceinline__ float fexp(float x) { return __expf(x); }
__device__ __forceinline__ float flog(float x) { return __logf(x); }

// =====================================================================
// Main fused kernel: one thread per (b, a). Streams all inputs once,
// computes BCE + CIoU + DFL contributions, block-reduces into a unique
// per-block slot (deterministic).
// DFL expectation (softmax @ [0..15]) is done with chained f32 WMMA
// 16x16x4 tiles through per-wave LDS staging.
// =====================================================================
__global__ __launch_bounds__(BDIM) void yolo_loss_main(
    const float* __restrict__ distri,      // [BANCH, 64]
    const float* __restrict__ scores,      // [BANCH, 80]
    const float* __restrict__ anchors,     // [NA, 2]
    const float* __restrict__ tboxes,      // [BANCH, 4]
    const float* __restrict__ tscores,     // [BANCH, 80]
    const unsigned char* __restrict__ fg,  // [BANCH] bool
    float* __restrict__ partial)           // [NBLOCKS, 4]
{
    __shared__ __align__(16) float lds_p[WAVES][32 * 64];  // softmax probs
    __shared__ __align__(16) float lds_d[WAVES][32][4];    // decoded distances
    __shared__ float red[WAVES][4];

    const int tid  = threadIdx.x;
    const int lane = tid & 31;
    const int w    = tid >> 5;
    const int idx  = blockIdx.x * BDIM + tid;     // always < BANCH (exact grid)
    const int a    = idx % NA;

    __builtin_prefetch(&scores[idx * NC], 0, 3);
    __builtin_prefetch(&tscores[idx * NC], 0, 3);

    const float ax = anchors[2 * a + 0];
    const float ay = anchors[2 * a + 1];
    const float4 tb = *(const float4*)&tboxes[idx * 4];

    // ---------------- per-side softmax + DFL cross-entropy ----------------
    float dflacc = 0.0f;
    #pragma unroll
    for (int s = 0; s < 4; ++s) {
        const float4* px = (const float4*)&distri[idx * 64 + s * 16];
        float4 q0 = px[0], q1 = px[1], q2 = px[2], q3 = px[3];
        float x[16] = {q0.x, q0.y, q0.z, q0.w, q1.x, q1.y, q1.z, q1.w,
                       q2.x, q2.y, q2.z, q2.w, q3.x, q3.y, q3.z, q3.w};
        float m = x[0];
        #pragma unroll
        for (int i = 1; i < 16; ++i) m = fmaxf(m, x[i]);
        float e[16], sum = 0.0f;
        #pragma unroll
        for (int i = 0; i < 16; ++i) { e[i] = fexp(x[i] - m); sum += e[i]; }
        const float inv = 1.0f / sum;
        const float lse = m + flog(sum);            // log-sum-exp: logp_i = x_i - lse

        // stage probabilities into this wave's LDS slice (row = lane)
        float* dst = &lds_p[w][lane * 64 + s * 16];
        #pragma unroll
        for (int i = 0; i < 4; ++i)
            *(float4*)(dst + 4 * i) = make_float4(e[4*i+0]*inv, e[4*i+1]*inv,
                                                  e[4*i+2]*inv, e[4*i+3]*inv);

        // DFL target distance for this side (ltrb), clipped to [0, 14.99]
        float td = (s == 0) ? (ax - tb.x) : (s == 1) ? (ay - tb.y)
                 : (s == 2) ? (tb.z - ax) : (tb.w - ay);
        td = fminf(fmaxf(td, 0.0f), (float)(REGMAX - 1) - 0.01f);
        const int   tl = (int)td;
        const float wl = (float)(tl + 1) - td;
        const float wr = 1.0f - wl;
        float xl = 0.0f, xr = 0.0f;
        #pragma unroll
        for (int i = 0; i < 16; ++i) {
            xl = (i == tl)     ? x[i] : xl;
            xr = (i == tl + 1) ? x[i] : xr;
        }
        // ce_l*wl + ce_r*wr = lse - (x[tl]*wl + x[tl+1]*wr)
        dflacc += lse - (xl * wl + xr * wr);
    }
    dflacc *= 0.25f;                                // mean over 4 sides

    // ---------------- WMMA: dist = P(16x16) @ proj via 4 chained K=4 tiles --
    asm volatile("" ::: "memory");                  // order LDS stores before loads
    {
        const int hi  = lane >> 4;                  // 0/1: which K pair this lane holds
        const int row = lane & 15;                  // M row within tile
        #pragma unroll
        for (int t = 0; t < 2; ++t) {               // 2 tiles of 16 anchors
            #pragma unroll
            for (int s = 0; s < 4; ++s) {           // 4 box sides
                v8f acc = {0.f, 0.f, 0.f, 0.f, 0.f, 0.f, 0.f, 0.f};
                const float* src =
                    &lds_p[w][(t * 16 + row) * 64 + s * 16 + hi * 2];
                #pragma unroll
                for (int c = 0; c < 4; ++c) {       // K chunks: bins 4c..4c+3
                    v2f av = *(const v2f*)(src + 4 * c);         // A: probs (doc layout)
                    float kb = (float)(4 * c + 2 * hi);
                    v2f bv = {kb, kb + 1.0f};                    // B: proj bcast along N
                    acc = __builtin_amdgcn_wmma_f32_16x16x4_f32(
                        false, av, false, bv, (short)0, acc, false, false);
                }
                if (row == 0) {                     // C/D layout: lane0 -> M0..7, lane16 -> M8..15
                    const int base = t * 16 + hi * 8;
                    #pragma unroll
                    for (int r = 0; r < 8; ++r) lds_d[w][base + r][s] = acc[r];
                }
            }
        }
    }
    asm volatile("" ::: "memory");
    const float4 dist = *(const float4*)&lds_d[w][lane][0];

    // ---------------- CIoU (exact reference formulation) ----------------
    const float b1x1 = ax - dist.x, b1y1 = ay - dist.y;
    const float b1x2 = ax + dist.z, b1y2 = ay + dist.w;
    const float b2x1 = tb.x, b2y1 = tb.y, b2x2 = tb.z, b2y2 = tb.w;
    const float w1 = b1x2 - b1x1, h1 = b1y2 - b1y1 + EPSF;
    const float w2 = b2x2 - b2x1, h2 = b2y2 - b2y1 + EPSF;
    const float iw = fmaxf(fminf(b1x2, b2x2) - fmaxf(b1x1, b2x1), 0.0f);
    const float ih = fmaxf(fminf(b1y2, b2y2) - fmaxf(b1y1, b2y1), 0.0f);
    const float inter = iw * ih;
    const float uni = w1 * h1 + w2 * h2 - inter + EPSF;
    const float iou = inter / uni;
    const float cw = fmaxf(b1x2, b2x2) - fminf(b1x1, b2x1);
    const float ch = fmaxf(b1y2, b2y2) - fminf(b1y1, b2y1);
    const float c2 = cw * cw + ch * ch + EPSF;
    const float dxc = b2x1 + b2x2 - b1x1 - b1x2;
    const float dyc = b2y1 + b2y2 - b1y1 - b1y2;
    const float rho2 = (dxc * dxc + dyc * dyc) * 0.25f;
    const float dva = atanf(w2 / h2) - atanf(w1 / h1);
    const float vv = (4.0f / (float)(M_PI * M_PI)) * dva * dva;
    const float alpha = vv / (vv - iou + (1.0f + EPSF));
    const float ciou = iou - (rho2 / c2 + vv * alpha);

    // ---------------- BCE-with-logits + target-score sum ----------------
    // bce = max(x,0) - x*t + log1p(exp(-|x|)); z=exp(-|x|) in (0,1] so
    // log(1+z) has no cancellation -> safe with native log.
    float bce = 0.0f, tsum = 0.0f;
    const float4* ps = (const float4*)&scores[idx * NC];
    const float4* pt = (const float4*)&tscores[idx * NC];
    #pragma unroll 5
    for (int c = 0; c < NC / 4; ++c) {
        const float4 xs = ps[c], ts = pt[c];
        bce += fmaxf(xs.x, 0.f) - xs.x * ts.x + flog(1.0f + fexp(-fabsf(xs.x)));
        bce += fmaxf(xs.y, 0.f) - xs.y * ts.y + flog(1.0f + fexp(-fabsf(xs.y)));
        bce += fmaxf(xs.z, 0.f) - xs.z * ts.z + flog(1.0f + fexp(-fabsf(xs.z)));
        bce += fmaxf(xs.w, 0.f) - xs.w * ts.w + flog(1.0f + fexp(-fabsf(xs.w)));
        tsum += ts.x + ts.y + ts.z + ts.w;
    }

    const float fgv = fg[idx] ? 1.0f : 0.0f;
    const float weight = tsum * fgv;
    float v0 = tsum;                       // sum(target_scores)
    float v1 = bce;                        // sum(bce)
    float v2 = (1.0f - ciou) * weight;     // sum((1-iou)*w)
    float v3 = dflacc * weight;            // sum(dfl*w)

    // ---------------- deterministic block reduction ----------------
    #pragma unroll
    for (int off = 16; off > 0; off >>= 1) {
        v0 += __shfl_down(v0, off, 32);
        v1 += __shfl_down(v1, off, 32);
        v2 += __shfl_down(v2, off, 32);
        v3 += __shfl_down(v3, off, 32);
    }
    if (lane == 0) { red[w][0] = v0; red[w][1] = v1; red[w][2] = v2; red[w][3] = v3; }
    __syncthreads();
    if (tid == 0) {
        float s0 = 0, s1 = 0, s2 = 0, s3 = 0;
        #pragma unroll
        for (int i = 0; i < WAVES; ++i) {
            s0 += red[i][0]; s1 += red[i][1]; s2 += red[i][2]; s3 += red[i][3];
        }
        float* out = &partial[blockIdx.x * 4];
        out[0] = s0; out[1] = s1; out[2] = s2; out[3] = s3;
    }
}

// =====================================================================
// Final fixed-order reduction over per-block partials -> scalar loss
// =====================================================================
__global__ __launch_bounds__(256) void yolo_loss_final(
    const float* __restrict__ partial, float* __restrict__ out)
{
    __shared__ float sh[256][4];
    const int t = threadIdx.x;
    float s[4] = {0.f, 0.f, 0.f, 0.f};
    for (int j = t; j < NBLOCKS; j += 256) {
        s[0] += partial[j * 4 + 0];
        s[1] += partial[j * 4 + 1];
        s[2] += partial[j * 4 + 2];
        s[3] += partial[j * 4 + 3];
    }
    sh[t][0] = s[0]; sh[t][1] = s[1]; sh[t][2] = s[2]; sh[t][3] = s[3];
    __syncthreads();
    for (int off = 128; off > 0; off >>= 1) {
        if (t < off) {
            sh[t][0] += sh[t + off][0];
            sh[t][1] += sh[t + off][1];
            sh[t][2] += sh[t + off][2];
            sh[t][3] += sh[t + off][3];
        }
        __syncthreads();
    }
    if (t == 0) {
        const float sts  = sh[0][0];
        const float sbce = sh[0][1];
        const float siou = sh[0][2];
        const float sdfl = sh[0][3];
        const float tss = fmaxf(sts, 1.0f);
        out[0] = (float)NB * (7.5f * siou + 0.5f * sbce + 1.5f * sdfl) / tss;
    }
}

extern "C" void kernel_launch(void* const* d_in, const int* in_sizes, int n_in,
                              void* d_out, int out_size, void* d_ws, size_t ws_size,
                              hipStream_t stream) {
    const float* pred_distri   = (const float*)d_in[0];
    const float* pred_scores   = (const float*)d_in[1];
    const float* anchor_points = (const float*)d_in[2];
    const float* target_bboxes = (const float*)d_in[3];
    const float* target_scores = (const float*)d_in[4];
    const unsigned char* fg    = (const unsigned char*)d_in[5];

    float* partial = (float*)d_ws;   // NBLOCKS*4 floats (33.6 KB) scratch
    float* out     = (float*)d_out;

    yolo_loss_main<<<NBLOCKS, BDIM, 0, stream>>>(
        pred_distri, pred_scores, anchor_points, target_bboxes,
        target_scores, fg, partial);
    yolo_loss_final<<<1, 256, 0, stream>>>(partial, out);
}